// ArccosHessianCalculator_42365557408121
// MI455X (gfx1250) — compile-verified
//
#include <hip/hip_runtime.h>

// Problem constants from the reference
#define NROWS 16384
#define DIN   512
#define DOUT  256
#define BSZ   1024

typedef __attribute__((ext_vector_type(2))) float v2f;
typedef __attribute__((ext_vector_type(8))) float v8f;

// ---------------------------------------------------------------------------
// Kernel 1: gather x rows for the 4 index sets into dense xg[4][BSZ][DIN]
// set 0 = ap (pos x1), 1 = p (pos x2), 2 = an (neg x1), 3 = n (neg x2)
// ---------------------------------------------------------------------------
__global__ void gather_rows_kernel(const float* __restrict__ x,
                                   const int* __restrict__ ap,
                                   const int* __restrict__ p,
                                   const int* __restrict__ an,
                                   const int* __restrict__ n,
                                   float* __restrict__ xg) {
    const int set = blockIdx.y;          // 0..3
    const int b   = blockIdx.x;          // 0..BSZ-1
    const int* idx = (set == 0) ? ap : (set == 1) ? p : (set == 2) ? an : n;
    const int row = idx[b];
    const float4* src = (const float4*)(x + (size_t)row * DIN);
    float4* dst = (float4*)(xg + ((size_t)set * BSZ + b) * DIN);
    dst[threadIdx.x] = src[threadIdx.x];           // 128 threads x float4 = 512 floats
}

// ---------------------------------------------------------------------------
// Kernel 2: z[set] = xg[set] @ W^T  via V_WMMA_F32_16X16X4_F32
//   z: [4][BSZ][DOUT], one 16x16 output tile per wave, K = DIN in steps of 4.
//   A[m][kk] = xg[m0+m][k+kk]        (contiguous float2 per lane)
//   B[kk][n] = W^T[k+kk][n0+n] = W[n0+n][k+kk]  (contiguous float2 per lane)
// ---------------------------------------------------------------------------
__global__ void zgemm_wmma_kernel(const float* __restrict__ xg,
                                  const float* __restrict__ W,
                                  float* __restrict__ z) {
    const int wave   = blockIdx.x * (blockDim.x >> 5) + (threadIdx.x >> 5);
    const int lane   = threadIdx.x & 31;
    const int half   = lane >> 4;       // which K-pair this lane holds
    const int lane16 = lane & 15;
    const int kb     = half * 2;

    const int tilesN = DOUT / 16;                  // 16
    const int tilesM = BSZ / 16;                   // 64
    const int tilesPerSet = tilesM * tilesN;       // 1024
    const int set = wave / tilesPerSet;            // 0..3 (grid sized exactly)
    const int rem = wave % tilesPerSet;
    const int m0  = (rem / tilesN) * 16;
    const int n0  = (rem % tilesN) * 16;

    const float* __restrict__ arow = xg + ((size_t)set * BSZ + m0 + lane16) * DIN + kb;
    const float* __restrict__ brow = W  + ((size_t)(n0 + lane16)) * DIN + kb;

    v8f acc = {0.f, 0.f, 0.f, 0.f, 0.f, 0.f, 0.f, 0.f};
#pragma unroll 8
    for (int k = 0; k < DIN; k += 4) {
        v2f a = *(const v2f*)(arow + k);           // 8B aligned: kb even, k % 4 == 0
        v2f b = *(const v2f*)(brow + k);
        acc = __builtin_amdgcn_wmma_f32_16x16x4_f32(
            /*neg_a=*/false, a, /*neg_b=*/false, b,
            /*c_mod=*/(short)0, acc, /*reuse_a=*/false, /*reuse_b=*/false);
    }

    float* __restrict__ zs = z + (size_t)set * BSZ * DOUT;
#pragma unroll
    for (int r = 0; r < 8; ++r)
        zs[(size_t)(m0 + r + 8 * half) * DOUT + n0 + lane16] = acc[r];
}

// ---------------------------------------------------------------------------
// Kernel 3: per (group, row) build the signed L rows (diag Hessian factors)
// and R rows (elementwise x products).
//   L: [6][BSZ][DOUT]   row t*BSZ+b :  t = g*3 + {0:d11, 1:-2*d12, 2:d22}, sign (+/-)
//   R: [6][BSZ][DIN]    row t*BSZ+b :  {x1*x1, x1*x2, x2*x2}
// ---------------------------------------------------------------------------
__global__ void prep_kernel(const float* __restrict__ xg,
                            const float* __restrict__ z,
                            float* __restrict__ L,
                            float* __restrict__ R) {
    const int g = blockIdx.x / BSZ;   // 0 = pos, 1 = neg
    const int b = blockIdx.x % BSZ;
    const int tid = threadIdx.x;      // 256 threads

    const float* __restrict__ z1 = z + ((size_t)(2 * g + 0) * BSZ + b) * DOUT;
    const float* __restrict__ z2 = z + ((size_t)(2 * g + 1) * BSZ + b) * DOUT;
    const float* __restrict__ x1 = xg + ((size_t)(2 * g + 0) * BSZ + b) * DIN;
    const float* __restrict__ x2 = xg + ((size_t)(2 * g + 1) * BSZ + b) * DIN;

    const float z1v = z1[tid];
    const float z2v = z2[tid];

    __shared__ float sh[3 * DOUT];
    sh[tid]            = z1v * z1v;
    sh[DOUT + tid]     = z2v * z2v;
    sh[2 * DOUT + tid] = z1v * z2v;
    __syncthreads();
    for (int off = 128; off > 0; off >>= 1) {
        if (tid < off) {
            sh[tid]            += sh[tid + off];
            sh[DOUT + tid]     += sh[DOUT + tid + off];
            sh[2 * DOUT + tid] += sh[2 * DOUT + tid + off];
        }
        __syncthreads();
    }
    const float s11 = sh[0];                // ||z1||^2
    const float s22 = sh[DOUT];             // ||z2||^2
    const float s12 = sh[2 * DOUT];         // <z1, z2>
    const float n1  = sqrtf(s11);
    const float n2  = sqrtf(s22);
    const float n12 = n1 * n2;
    const float c   = s12 / n12;            // cosine

    const float u = z1v / n1;
    const float v = z2v / n2;
    const float uv = u * v;
    const float d11 = (c + 2.f * uv - 3.f * c * u * u) / s11;
    const float d12 = (-1.f + u * u + v * v - c * uv) / n12;
    const float d22 = (c + 2.f * uv - 3.f * c * v * v) / s22;

    const float sgn = (g == 0) ? 1.f : -1.f;
    const int t0 = g * 3;
    L[((size_t)(t0 + 0) * BSZ + b) * DOUT + tid] = sgn * d11;
    L[((size_t)(t0 + 1) * BSZ + b) * DOUT + tid] = sgn * (-2.f) * d12;
    L[((size_t)(t0 + 2) * BSZ + b) * DOUT + tid] = sgn * d22;

#pragma unroll
    for (int i = tid; i < DIN; i += 256) {
        const float a1 = x1[i];
        const float a2 = x2[i];
        R[((size_t)(t0 + 0) * BSZ + b) * DIN + i] = a1 * a1;
        R[((size_t)(t0 + 1) * BSZ + b) * DIN + i] = a1 * a2;
        R[((size_t)(t0 + 2) * BSZ + b) * DIN + i] = a2 * a2;
    }
}

// ---------------------------------------------------------------------------
// Kernel 4: out(DOUT x DIN) = L^T (DOUT x 6144) @ R (6144 x DIN) via WMMA f32.
//   A[m][kk] = L[k+kk][m0+m]   (column of row-major L, stride DOUT)
//   B[kk][n] = R[k+kk][n0+n]   (row-major, stride DIN)
// ---------------------------------------------------------------------------
__global__ void outer_gemm_wmma_kernel(const float* __restrict__ L,
                                       const float* __restrict__ R,
                                       float* __restrict__ out) {
    const int wave   = blockIdx.x * (blockDim.x >> 5) + (threadIdx.x >> 5);
    const int lane   = threadIdx.x & 31;
    const int half   = lane >> 4;
    const int lane16 = lane & 15;
    const int kb     = half * 2;

    const int tilesN = DIN / 16;                   // 32
    const int m0 = (wave / tilesN) * 16;           // 0..240
    const int n0 = (wave % tilesN) * 16;           // 0..496
    const int KTOT = 6 * BSZ;                      // 6144

    const float* __restrict__ lp = L + (size_t)kb * DOUT + m0 + lane16;
    const float* __restrict__ rp = R + (size_t)kb * DIN + n0 + lane16;

    v8f acc = {0.f, 0.f, 0.f, 0.f, 0.f, 0.f, 0.f, 0.f};
#pragma unroll 4
    for (int k = 0; k < KTOT; k += 4) {
        v2f a; a.x = lp[0]; a.y = lp[DOUT];
        v2f b; b.x = rp[0]; b.y = rp[DIN];
        acc = __builtin_amdgcn_wmma_f32_16x16x4_f32(
            false, a, false, b, (short)0, acc, false, false);
        lp += 4 * DOUT;
        rp += 4 * DIN;
    }

#pragma unroll
    for (int r = 0; r < 8; ++r)
        out[(size_t)(m0 + r + 8 * half) * DIN + n0 + lane16] = acc[r];
}

// ---------------------------------------------------------------------------
// Launch: gather -> z GEMM -> prep -> final accumulation GEMM (stream-ordered)
// Workspace layout (floats):
//   xg : 4*BSZ*DIN   = 2,097,152
//   z  : 4*BSZ*DOUT  = 1,048,576
//   L  : 6*BSZ*DOUT  = 1,572,864
//   R  : 6*BSZ*DIN   = 3,145,728          total ~30 MB
// ---------------------------------------------------------------------------
extern "C" void kernel_launch(void* const* d_in, const int* in_sizes, int n_in,
                              void* d_out, int out_size, void* d_ws, size_t ws_size,
                              hipStream_t stream) {
    const float* x  = (const float*)d_in[0];
    const float* W  = (const float*)d_in[1];
    const int*   ap = (const int*)d_in[2];
    const int*   p  = (const int*)d_in[3];
    const int*   an = (const int*)d_in[4];
    const int*   n  = (const int*)d_in[5];
    float* out = (float*)d_out;

    float* xg = (float*)d_ws;
    float* z  = xg + (size_t)4 * BSZ * DIN;
    float* L  = z  + (size_t)4 * BSZ * DOUT;
    float* R  = L  + (size_t)6 * BSZ * DOUT;

    dim3 gblk(BSZ, 4);
    gather_rows_kernel<<<gblk, 128, 0, stream>>>(x, ap, p, an, n, xg);

    // 4 sets * 64 * 16 = 4096 tiles; 8 waves / 256-thread block -> 512 blocks
    zgemm_wmma_kernel<<<512, 256, 0, stream>>>(xg, W, z);

    prep_kernel<<<2 * BSZ, 256, 0, stream>>>(xg, z, L, R);

    // (256/16)*(512/16) = 512 tiles; 8 waves/block -> 64 blocks
    outer_gemm_wmma_kernel<<<64, 256, 0, stream>>>(L, R, out);
}